// OrthogonalSylvesterVAE_55783035240477
// MI455X (gfx1250) — compile-verified
//
#include <hip/hip_runtime.h>
#include <math.h>

// ---------------------------------------------------------------------------
// OrthogonalSylvesterVAE forward for MI455X (gfx1250, wave32, WMMA).
// Heavy compute (Bjorck orthogonalization ~45 GFLOP + amortization GEMMs
// ~5 GFLOP) runs on V_WMMA_F32_16X16X4_F32; everything else is tiny VALU work.
// ---------------------------------------------------------------------------

typedef float v2f __attribute__((ext_vector_type(2)));
typedef float v8f __attribute__((ext_vector_type(8)));

// 8-arg pattern per probe doc: (neg_a, A, neg_b, B, c_mod, C, reuse_a, reuse_b)
#define WMMA_F32(a, b, c) \
  __builtin_amdgcn_wmma_f32_16x16x4_f32(false, (a), false, (b), (short)0, (c), false, false)

#define BN_EPS 1e-5f

// ------------------------------ GEMM (WMMA) --------------------------------
// C[M,N] = A[M,K] @ W[N,K]^T + bias.  One wave computes a 16x32 C strip (two
// 16x16 tiles sharing one A fragment); block = 4 waves -> 128 columns.
// Wave-uniform fast path when all 32 columns are in bounds (no masking at
// all); otherwise branchless clamp+mask.  Both paths are software-pipelined:
// fragments for step k+1 are loaded before the WMMAs of step k issue, so
// global_load_b64s stay in flight behind the matrix ops.
// Fragment layouts per CDNA5 ISA 7.12.2:
//   A (16x4 MxK): lane L -> m = L&15, vgpr j -> k = 2*(L>>4)+j
//   B (4x16 KxN): assumed symmetric: lane L -> n = L&15, vgpr j -> k = 2*(L>>4)+j
//   C/D (16x16):  lane L, vgpr v -> row = v + 8*(L>>4), col = L&15
__global__ __launch_bounds__(128) void gemm_bias_wmma(
    const float* __restrict__ A, int lda,
    const float* __restrict__ W, int ldw,
    const float* __restrict__ bias,
    float* __restrict__ C, int ldc,
    int M, int N, int K) {
  int lane = threadIdx.x & 31;
  int wave = threadIdx.x >> 5;
  int j0 = (blockIdx.x * 4 + wave) * 32;
  int i0 = blockIdx.y * 16;
  if (j0 >= N) return;  // wave-uniform exit
  int half = lane >> 4;
  int l15 = lane & 15;
  int arow = i0 + l15;

  int w0 = j0 + l15;
  int w1 = j0 + 16 + l15;
  int w0s = (w0 < N) ? w0 : (N - 1);
  int w1s = (w1 < N) ? w1 : (N - 1);
  float m0 = (w0 < N) ? 1.f : 0.f;
  float m1 = (w1 < N) ? 1.f : 0.f;

  const float* Ap = A + (size_t)arow * lda + 2 * half;
  const float* W0 = W + (size_t)w0s * ldw + 2 * half;
  const float* W1 = W + (size_t)w1s * ldw + 2 * half;

  v8f c0 = {0.f, 0.f, 0.f, 0.f, 0.f, 0.f, 0.f, 0.f};
  v8f c1 = {0.f, 0.f, 0.f, 0.f, 0.f, 0.f, 0.f, 0.f};

  int Kmain = K >> 2;
  if (Kmain > 0) {
    v2f av, b0, b1, an, bn0, bn1;
    // prologue: load step 0
    av.x = Ap[0]; av.y = Ap[1];
    b0.x = W0[0]; b0.y = W0[1];
    b1.x = W1[0]; b1.y = W1[1];
    if (j0 + 32 <= N) {
      // ---- fast path: no masks at all ----
      for (int kk = 0; kk < Kmain - 1; ++kk) {
        an.x = Ap[4];  an.y = Ap[5];    // prefetch step kk+1
        bn0.x = W0[4]; bn0.y = W0[5];
        bn1.x = W1[4]; bn1.y = W1[5];
        c0 = WMMA_F32(av, b0, c0);      // overlaps with loads above
        c1 = WMMA_F32(av, b1, c1);
        av = an; b0 = bn0; b1 = bn1;
        Ap += 4; W0 += 4; W1 += 4;
      }
      c0 = WMMA_F32(av, b0, c0);
      c1 = WMMA_F32(av, b1, c1);
    } else {
      // ---- masked path (N edge): clamp rows + 0/1 multiply ----
      b0.x *= m0; b0.y *= m0;
      b1.x *= m1; b1.y *= m1;
      for (int kk = 0; kk < Kmain - 1; ++kk) {
        an.x = Ap[4];       an.y = Ap[5];
        bn0.x = W0[4] * m0; bn0.y = W0[5] * m0;
        bn1.x = W1[4] * m1; bn1.y = W1[5] * m1;
        c0 = WMMA_F32(av, b0, c0);
        c1 = WMMA_F32(av, b1, c1);
        av = an; b0 = bn0; b1 = bn1;
        Ap += 4; W0 += 4; W1 += 4;
      }
      c0 = WMMA_F32(av, b0, c0);
      c1 = WMMA_F32(av, b1, c1);
    }
  }
  if (K & 3) {  // masked tail step, indices clamped in-bounds
    int kb = (Kmain << 2) + 2 * half;
    float km0 = (kb < K) ? 1.f : 0.f;
    float km1 = (kb + 1 < K) ? 1.f : 0.f;
    int o0 = (kb < K) ? kb : (K - 1);
    int o1 = (kb + 1 < K) ? (kb + 1) : (K - 1);
    const float* Arow = A + (size_t)arow * lda;
    const float* Wr0 = W + (size_t)w0s * ldw;
    const float* Wr1 = W + (size_t)w1s * ldw;
    v2f av, b0, b1;
    av.x = Arow[o0] * km0;
    av.y = Arow[o1] * km1;
    b0.x = Wr0[o0] * m0;
    b0.y = Wr0[o1] * m0;
    b1.x = Wr1[o0] * m1;
    b1.y = Wr1[o1] * m1;
    c0 = WMMA_F32(av, b0, c0);
    c1 = WMMA_F32(av, b1, c1);
  }

  int col0 = j0 + l15;
  int col1 = j0 + 16 + l15;
  if (col0 < N) {
    float bvv = bias ? bias[col0] : 0.f;
    for (int v = 0; v < 8; ++v) {
      int row = i0 + v + 8 * half;
      C[(size_t)row * ldc + col0] = c0[v] + bvv;
    }
  }
  if (col1 < N) {
    float bvv = bias ? bias[col1] : 0.f;
    for (int v = 0; v < 8; ++v) {
      int row = i0 + v + 8 * half;
      C[(size_t)row * ldc + col1] = c1[v] + bvv;
    }
  }
}

// ------------------------- Bjorck orthogonalization ------------------------
// q: [8192 rows of 576] (= [B*F, Z=24, M=24]), already row-normalized.
// One wave owns one matrix in a zero-padded 32x32 LDS tile (stride 33).
// Per step: G = A^T A (WMMA, Kloop over Z=24), T = 1.5I - 0.5G -> LDS,
// A <- A @ T (WMMA, Kloop over M=24).  100 steps.
__global__ __launch_bounds__(128) void bjorck_kernel(float* __restrict__ q) {
  const int BS = 33;
  __shared__ float lds[4 * 2 * 32 * 33];  // 33.8 KB: 4 waves x (A,T)
  int lane = threadIdx.x & 31;
  int wave = threadIdx.x >> 5;
  int half = lane >> 4;
  int l15 = lane & 15;
  size_t n = (size_t)blockIdx.x * 4 + wave;
  float* Ab = &lds[wave * 2 * 32 * BS];
  float* Tb = Ab + 32 * BS;
  for (int i = lane; i < 32 * BS; i += 32) Ab[i] = 0.f;
  __syncthreads();
  float* g = q + n * 576;
  for (int i = lane; i < 576; i += 32) {
    int z = i / 24, m = i - z * 24;
    Ab[z * BS + m] = g[i];
  }
  __syncthreads();
  for (int step = 0; step < 100; ++step) {
    // T = 1.5I - 0.5 * A^T A   (A^T tile: element[i][k] = Ab[k][i])
    for (int ti = 0; ti < 2; ++ti)
      for (int tj = 0; tj < 2; ++tj) {
        v8f acc = {0.f, 0.f, 0.f, 0.f, 0.f, 0.f, 0.f, 0.f};
        for (int kk = 0; kk < 6; ++kk) {   // K = Z = 24 (padded rows are 0)
          int kb = kk * 4 + 2 * half;
          v2f av, bv;
          av.x = Ab[kb * BS + ti * 16 + l15];
          av.y = Ab[(kb + 1) * BS + ti * 16 + l15];
          bv.x = Ab[kb * BS + tj * 16 + l15];
          bv.y = Ab[(kb + 1) * BS + tj * 16 + l15];
          acc = WMMA_F32(av, bv, acc);
        }
        for (int v = 0; v < 8; ++v) {
          int r = ti * 16 + v + 8 * half;
          int cc = tj * 16 + l15;
          Tb[r * BS + cc] = -0.5f * acc[v] + ((r == cc) ? 1.5f : 0.f);
        }
      }
    __syncthreads();
    // A <- A @ T
    v8f an[2][2];
    for (int tz = 0; tz < 2; ++tz)
      for (int tm = 0; tm < 2; ++tm) {
        v8f acc = {0.f, 0.f, 0.f, 0.f, 0.f, 0.f, 0.f, 0.f};
        for (int kk = 0; kk < 6; ++kk) {  // K = M = 24
          int kb = kk * 4 + 2 * half;
          v2f av, bv;
          av.x = Ab[(tz * 16 + l15) * BS + kb];   // contiguous pair -> ds_load_b64
          av.y = Ab[(tz * 16 + l15) * BS + kb + 1];
          bv.x = Tb[kb * BS + tm * 16 + l15];
          bv.y = Tb[(kb + 1) * BS + tm * 16 + l15];
          acc = WMMA_F32(av, bv, acc);
        }
        an[tz][tm] = acc;
      }
    __syncthreads();
    for (int tz = 0; tz < 2; ++tz)
      for (int tm = 0; tm < 2; ++tm)
        for (int v = 0; v < 8; ++v)
          Ab[(tz * 16 + v + 8 * half) * BS + tm * 16 + l15] = an[tz][tm][v];
    __syncthreads();
  }
  for (int i = lane; i < 576; i += 32) {
    int z = i / 24, m = i - z * 24;
    g[i] = Ab[z * BS + m];
  }
}

// row L2-normalization for q: 8192 rows of 576
__global__ void norm_rows(float* __restrict__ q) {
  int n = blockIdx.x, t = threadIdx.x;  // 64 threads
  __shared__ float red[64];
  float* r = q + (size_t)n * 576;
  float s = 0.f;
  for (int i = t; i < 576; i += 64) { float v = r[i]; s += v * v; }
  red[t] = s; __syncthreads();
  for (int off = 32; off > 0; off >>= 1) {
    if (t < off) red[t] += red[t + off];
    __syncthreads();
  }
  float inv = 1.0f / sqrtf(red[0]);
  for (int i = t; i < 576; i += 64) r[i] *= inv;
}

// ---------------------------- BN (training mode) ---------------------------
__global__ void bn_stats(const float* __restrict__ x, float* __restrict__ stats,
                         int N, int C, int L) {
  int c = blockIdx.x, t = threadIdx.x;
  __shared__ float s1[256], s2[256];
  float a = 0.f, b = 0.f;
  int total = N * L;
  for (int i = t; i < total; i += 256) {
    int nn = i / L, l = i - nn * L;
    float v = x[((size_t)nn * C + c) * L + l];
    a += v; b += v * v;
  }
  s1[t] = a; s2[t] = b; __syncthreads();
  for (int off = 128; off > 0; off >>= 1) {
    if (t < off) { s1[t] += s1[t + off]; s2[t] += s2[t + off]; }
    __syncthreads();
  }
  if (t == 0) {
    float mu = s1[0] / (float)total;
    stats[c] = mu;
    stats[C + c] = s2[0] / (float)total - mu * mu;
  }
}

__global__ void bn_apply(const float* __restrict__ x, const float* __restrict__ stats,
                         const float* __restrict__ gam, const float* __restrict__ bet,
                         float* __restrict__ out, int N, int C, int L,
                         int out_stride, int relu) {
  size_t i = (size_t)blockIdx.x * 256 + threadIdx.x;
  size_t total = (size_t)N * C * L;
  if (i >= total) return;
  int l = (int)(i % L);
  int cc = (int)((i / L) % C);
  int nn = (int)(i / ((size_t)C * L));
  float v = x[i];
  float y = (v - stats[cc]) * rsqrtf(stats[C + cc] + BN_EPS) * gam[cc] + bet[cc];
  if (relu) y = fmaxf(y, 0.f);
  out[(size_t)nn * out_stride + (size_t)cc * L + l] = y;
}

// ------------------------------- conv / deconv -----------------------------
__global__ void conv1_kernel(const float* __restrict__ x, const float* __restrict__ w,
                             const float* __restrict__ bias, float* __restrict__ out) {
  int idx = blockIdx.x * 256 + threadIdx.x;  // 2048*32*30
  if (idx >= 2048 * 32 * 30) return;
  int oh = idx % 30;
  int oc = (idx / 30) % 32;
  int b = idx / (30 * 32);
  float s = bias[oc];
  for (int kh = 0; kh < 3; ++kh)
    for (int kw = 0; kw < 4; ++kw)
      s += x[b * 128 + (oh + kh) * 4 + kw] * w[oc * 12 + kh * 4 + kw];
  out[idx] = s;
}

__global__ void conv1d_valid(const float* __restrict__ x, const float* __restrict__ w,
                             const float* __restrict__ bias, float* __restrict__ out,
                             int B_, int IC, int IL, int OC, int OL, int KH) {
  int idx = blockIdx.x * 256 + threadIdx.x;
  int total = B_ * OC * OL;
  if (idx >= total) return;
  int o = idx % OL;
  int oc = (idx / OL) % OC;
  int b = idx / (OL * OC);
  float s = bias[oc];
  for (int ic = 0; ic < IC; ++ic)
    for (int kh = 0; kh < KH; ++kh)
      s += x[((size_t)b * IC + ic) * IL + o + kh] * w[((size_t)oc * IC + ic) * KH + kh];
  out[idx] = s;
}

// ConvTranspose2d (stride 1, pad 0, kw==1), weight [IC][OC][KH]
__global__ void deconv1d(const float* __restrict__ x, int in_row_stride, int IL,
                         const float* __restrict__ w, const float* __restrict__ bias,
                         float* __restrict__ out, int B_, int IC, int OC, int OL, int KH) {
  int idx = blockIdx.x * 256 + threadIdx.x;
  int total = B_ * OC * OL;
  if (idx >= total) return;
  int o = idx % OL;
  int oc = (idx / OL) % OC;
  int b = idx / (OL * OC);
  float s = bias[oc];
  for (int ic = 0; ic < IC; ++ic)
    for (int kh = 0; kh < KH; ++kh) {
      int ih = o - kh;
      if (ih >= 0 && ih < IL)
        s += x[(size_t)b * in_row_stride + ic * IL + ih] * w[((size_t)ic * OC + oc) * KH + kh];
    }
  out[idx] = s;
}

// dec3: ConvTranspose2d in=32,out=1,k=(3,4), input [B,32,30,1] -> [B,1,32,4]
__global__ void dec3_kernel(const float* __restrict__ e2, const float* __restrict__ w,
                            const float* __restrict__ bias, float* __restrict__ out) {
  int idx = blockIdx.x * 256 + threadIdx.x;  // 2048*32*4
  if (idx >= 2048 * 128) return;
  int ow = idx & 3;
  int oh = (idx >> 2) % 32;
  int b = idx >> 7;
  float s = bias[0];
  for (int ic = 0; ic < 32; ++ic)
    for (int kh = 0; kh < 3; ++kh) {
      int ih = oh - kh;
      if (ih >= 0 && ih < 30)
        s += e2[((size_t)b * 32 + ic) * 30 + ih] * w[ic * 12 + kh * 4 + ow];
    }
  out[idx] = s;
}

// --------------------------------- misc ------------------------------------
__global__ void set_y_kernel(float* __restrict__ feat, const float* __restrict__ y) {
  int b = blockIdx.x * 256 + threadIdx.x;
  if (b < 2048) feat[(size_t)b * 161 + 160] = y[b];
}

__global__ void z0_kernel(const float* __restrict__ zmu, const float* __restrict__ zlv,
                          const float* __restrict__ eps, float* __restrict__ z0_out,
                          float* __restrict__ zk_ws) {
  int i = blockIdx.x * 256 + threadIdx.x;
  if (i >= 2048 * 24) return;
  float v = zmu[i] + eps[i] * expf(0.5f * zlv[i]);
  z0_out[i] = v;
  zk_ws[i] = v;
}

// ------------------------------- Sylvester flows ---------------------------
// One wave per batch element; F=4 flows chained in-kernel.
// r1[i][j] = fulld[b][(i*24+j)*4+k] (i<j), tanh(diag1) on diag, 0 below.
// r2[i][j] = fulld[b][(j*24+i)*4+k] (i<j), tanh(diag2) on diag, 0 below.
__global__ void flow_kernel(const float* __restrict__ qort, const float* __restrict__ fulld,
                            const float* __restrict__ d1, const float* __restrict__ d2,
                            const float* __restrict__ bfl, float* __restrict__ zk_ws,
                            float* __restrict__ out_zk, float* __restrict__ out_ld) {
  int b = blockIdx.x, t = threadIdx.x;  // 32 threads
  __shared__ float Qs[576], zs[24], us[24], hs[24], wv[24], ldacc[24];
  if (t < 24) { zs[t] = zk_ws[b * 24 + t]; ldacc[t] = 0.f; }
  __syncthreads();
  for (int k = 0; k < 4; ++k) {
    const float* Q = qort + ((size_t)b * 4 + k) * 576;
    for (int i = t; i < 576; i += 32) Qs[i] = Q[i];
    __syncthreads();
    if (t < 24) {  // u = Q^T z
      float s = 0.f;
      for (int z = 0; z < 24; ++z) s += zs[z] * Qs[z * 24 + t];
      us[t] = s;
    }
    __syncthreads();
    if (t < 24) {  // h = tanh(r2 Q^T z + b); logdet diagonal
      int nn = t;
      float s = bfl[b * 96 + nn * 4 + k] + us[nn] * tanhf(d2[b * 96 + nn * 4 + k]);
      for (int m = nn + 1; m < 24; ++m)
        s += us[m] * fulld[(size_t)b * 2304 + (m * 24 + nn) * 4 + k];
      float h = tanhf(s);
      hs[nn] = h;
      float r1nn = tanhf(d1[b * 96 + nn * 4 + k]);
      float r2nn = tanhf(d2[b * 96 + nn * 4 + k]);
      float dj = r1nn * r2nn * (1.f - h * h) + 1.f;
      ldacc[nn] += logf(fabsf(dj));
    }
    __syncthreads();
    if (t < 24) {  // w = r1 h
      int m = t;
      float s = tanhf(d1[b * 96 + m * 4 + k]) * hs[m];
      for (int nn = m + 1; nn < 24; ++nn)
        s += fulld[(size_t)b * 2304 + (m * 24 + nn) * 4 + k] * hs[nn];
      wv[m] = s;
    }
    __syncthreads();
    float znew = 0.f;
    if (t < 24) {  // z' = z + Q w
      float s = zs[t];
      for (int m = 0; m < 24; ++m) s += Qs[t * 24 + m] * wv[m];
      znew = s;
    }
    __syncthreads();
    if (t < 24) zs[t] = znew;
    __syncthreads();
  }
  if (t < 24) { out_zk[b * 24 + t] = zs[t]; zk_ws[b * 24 + t] = zs[t]; }
  if (t == 0) {
    float s = 0.f;
    for (int i = 0; i < 24; ++i) s += ldacc[i];
    out_ld[b] = s;
  }
}

// ------------------------------ host launcher ------------------------------
extern "C" void kernel_launch(void* const* d_in, const int* in_sizes, int n_in,
                              void* d_out, int out_size, void* d_ws, size_t ws_size,
                              hipStream_t stream) {
  (void)in_sizes; (void)n_in; (void)out_size; (void)ws_size;
  const float* x       = (const float*)d_in[0];
  const float* y       = (const float*)d_in[1];
  const float* eps     = (const float*)d_in[2];
  const float* conv1_w = (const float*)d_in[3];
  const float* conv1_b = (const float*)d_in[4];
  const float* bnc1_g  = (const float*)d_in[5];
  const float* bnc1_b  = (const float*)d_in[6];
  const float* conv2_w = (const float*)d_in[7];
  const float* conv2_b = (const float*)d_in[8];
  const float* bnc2_g  = (const float*)d_in[9];
  const float* bnc2_b  = (const float*)d_in[10];
  const float* conv3_w = (const float*)d_in[11];
  const float* conv3_b = (const float*)d_in[12];
  const float* bnc3_g  = (const float*)d_in[13];
  const float* bnc3_b  = (const float*)d_in[14];
  const float* dense1_w = (const float*)d_in[15];
  const float* dense1_b = (const float*)d_in[16];
  const float* bnd1_g   = (const float*)d_in[17];
  const float* bnd1_b   = (const float*)d_in[18];
  const float* qmean_w  = (const float*)d_in[19];
  const float* qmean_b  = (const float*)d_in[20];
  const float* qlogv_w  = (const float*)d_in[21];
  const float* qlogv_b  = (const float*)d_in[22];
  const float* amd_w    = (const float*)d_in[23];
  const float* amd_b    = (const float*)d_in[24];
  const float* amd1_w   = (const float*)d_in[25];
  const float* amd1_b   = (const float*)d_in[26];
  const float* amd2_w   = (const float*)d_in[27];
  const float* amd2_b   = (const float*)d_in[28];
  const float* amq_w    = (const float*)d_in[29];
  const float* amq_b    = (const float*)d_in[30];
  const float* amb_w    = (const float*)d_in[31];
  const float* amb_b    = (const float*)d_in[32];
  const float* dense3_w = (const float*)d_in[33];
  const float* dense3_b = (const float*)d_in[34];
  const float* bnd3_g   = (const float*)d_in[35];
  const float* bnd3_b   = (const float*)d_in[36];
  const float* dense4_w = (const float*)d_in[37];
  const float* dense4_b = (const float*)d_in[38];
  const float* bnd4_g   = (const float*)d_in[39];
  const float* bnd4_b   = (const float*)d_in[40];
  const float* dec1_w   = (const float*)d_in[41];
  const float* dec1_b   = (const float*)d_in[42];
  const float* bne1_g   = (const float*)d_in[43];
  const float* bne1_b   = (const float*)d_in[44];
  const float* dec2_w   = (const float*)d_in[45];
  const float* dec2_b   = (const float*)d_in[46];
  const float* bne2_g   = (const float*)d_in[47];
  const float* bne2_b   = (const float*)d_in[48];
  const float* dec3_w   = (const float*)d_in[49];
  const float* dec3_b   = (const float*)d_in[50];

  float* out = (float*)d_out;
  float* ws  = (float*)d_ws;

  // workspace layout (floats); encoder buffers reused by the decoder
  const size_t H1 = 0;                  // [2048,32,30] -> later E2
  const size_t H2 = 1966080;            // [2048,16,26] -> later E1
  const size_t H3 = 2818048;            // [2048,8,20]
  const size_t FEAT = 3145728;          // [2048,161]   -> later D4
  const size_t HD1 = 3475456;           // [2048,256]   -> later D3
  const size_t FULLD = 3999744;         // [2048,2304]
  const size_t D1B = 8718336;           // [2048,96]
  const size_t D2B = 8914944;           // [2048,96]
  const size_t QB = 9111552;            // [2048,2304] (becomes q_ortho in place)
  const size_t BFB = 13830144;          // [2048,96]
  const size_t ZKB = 14026752;          // [2048,24]
  const size_t STATS = 14075904;        // [512]

  // output layout (floats): x_dec, z_mu, z_logvar, log_det_j, z0, zk
  const size_t XDEC = 0;
  const size_t ZMU = 262144;
  const size_t ZLV = 311296;
  const size_t LDJ = 360448;
  const size_t Z0O = 362496;
  const size_t ZKO = 411648;

  // ---- encoder ----
  conv1_kernel<<<7680, 256, 0, stream>>>(x, conv1_w, conv1_b, ws + H1);
  bn_stats<<<32, 256, 0, stream>>>(ws + H1, ws + STATS, 2048, 32, 30);
  bn_apply<<<7680, 256, 0, stream>>>(ws + H1, ws + STATS, bnc1_g, bnc1_b,
                                     ws + H1, 2048, 32, 30, 960, 1);
  conv1d_valid<<<3328, 256, 0, stream>>>(ws + H1, conv2_w, conv2_b, ws + H2,
                                         2048, 32, 30, 16, 26, 5);
  bn_stats<<<16, 256, 0, stream>>>(ws + H2, ws + STATS, 2048, 16, 26);
  bn_apply<<<3328, 256, 0, stream>>>(ws + H2, ws + STATS, bnc2_g, bnc2_b,
                                     ws + H2, 2048, 16, 26, 416, 1);
  conv1d_valid<<<1280, 256, 0, stream>>>(ws + H2, conv3_w, conv3_b, ws + H3,
                                         2048, 16, 26, 8, 20, 7);
  bn_stats<<<8, 256, 0, stream>>>(ws + H3, ws + STATS, 2048, 8, 20);
  bn_apply<<<1280, 256, 0, stream>>>(ws + H3, ws + STATS, bnc3_g, bnc3_b,
                                     ws + FEAT, 2048, 8, 20, 161, 1);  // strided into feat
  set_y_kernel<<<8, 256, 0, stream>>>(ws + FEAT, y);

  // ---- dense1 + BN ----
  gemm_bias_wmma<<<dim3(2, 128), 128, 0, stream>>>(ws + FEAT, 161, dense1_w, 161,
                                                   dense1_b, ws + HD1, 256, 2048, 256, 161);
  bn_stats<<<256, 256, 0, stream>>>(ws + HD1, ws + STATS, 2048, 256, 1);
  bn_apply<<<2048, 256, 0, stream>>>(ws + HD1, ws + STATS, bnd1_g, bnd1_b,
                                     ws + HD1, 2048, 256, 1, 256, 1);

  // ---- amortization heads ----
  gemm_bias_wmma<<<dim3(1, 128), 128, 0, stream>>>(ws + HD1, 256, qmean_w, 256,
                                                   qmean_b, out + ZMU, 24, 2048, 24, 256);
  gemm_bias_wmma<<<dim3(1, 128), 128, 0, stream>>>(ws + HD1, 256, qlogv_w, 256,
                                                   qlogv_b, out + ZLV, 24, 2048, 24, 256);
  gemm_bias_wmma<<<dim3(18, 128), 128, 0, stream>>>(ws + HD1, 256, amd_w, 256,
                                                    amd_b, ws + FULLD, 2304, 2048, 2304, 256);
  gemm_bias_wmma<<<dim3(1, 128), 128, 0, stream>>>(ws + HD1, 256, amd1_w, 256,
                                                   amd1_b, ws + D1B, 96, 2048, 96, 256);
  gemm_bias_wmma<<<dim3(1, 128), 128, 0, stream>>>(ws + HD1, 256, amd2_w, 256,
                                                   amd2_b, ws + D2B, 96, 2048, 96, 256);
  gemm_bias_wmma<<<dim3(18, 128), 128, 0, stream>>>(ws + HD1, 256, amq_w, 256,
                                                    amq_b, ws + QB, 2304, 2048, 2304, 256);
  gemm_bias_wmma<<<dim3(1, 128), 128, 0, stream>>>(ws + HD1, 256, amb_w, 256,
                                                   amb_b, ws + BFB, 96, 2048, 96, 256);

  // ---- Bjorck orthogonalization (the hot loop) ----
  norm_rows<<<8192, 64, 0, stream>>>(ws + QB);
  bjorck_kernel<<<2048, 128, 0, stream>>>(ws + QB);

  // ---- reparameterize + flows ----
  z0_kernel<<<192, 256, 0, stream>>>(out + ZMU, out + ZLV, eps, out + Z0O, ws + ZKB);
  flow_kernel<<<2048, 32, 0, stream>>>(ws + QB, ws + FULLD, ws + D1B, ws + D2B,
                                       ws + BFB, ws + ZKB, out + ZKO, out + LDJ);

  // ---- decoder ----
  gemm_bias_wmma<<<dim3(2, 128), 128, 0, stream>>>(ws + ZKB, 24, dense3_w, 24,
                                                   dense3_b, ws + HD1, 256, 2048, 256, 24);
  bn_stats<<<256, 256, 0, stream>>>(ws + HD1, ws + STATS, 2048, 256, 1);
  bn_apply<<<2048, 256, 0, stream>>>(ws + HD1, ws + STATS, bnd3_g, bnd3_b,
                                     ws + HD1, 2048, 256, 1, 256, 1);
  gemm_bias_wmma<<<dim3(2, 128), 128, 0, stream>>>(ws + HD1, 256, dense4_w, 256,
                                                   dense4_b, ws + FEAT, 161, 2048, 161, 256);
  bn_stats<<<161, 256, 0, stream>>>(ws + FEAT, ws + STATS, 2048, 161, 1);
  bn_apply<<<1288, 256, 0, stream>>>(ws + FEAT, ws + STATS, bnd4_g, bnd4_b,
                                     ws + FEAT, 2048, 161, 1, 161, 1);
  // d4[:, :160] viewed as [B,8,20]; read strided (row stride 161)
  deconv1d<<<3328, 256, 0, stream>>>(ws + FEAT, 161, 20, dec1_w, dec1_b, ws + H2,
                                     2048, 8, 16, 26, 7);
  bn_stats<<<16, 256, 0, stream>>>(ws + H2, ws + STATS, 2048, 16, 26);
  bn_apply<<<3328, 256, 0, stream>>>(ws + H2, ws + STATS, bne1_g, bne1_b,
                                     ws + H2, 2048, 16, 26, 416, 1);
  deconv1d<<<7680, 256, 0, stream>>>(ws + H2, 416, 26, dec2_w, dec2_b, ws + H1,
                                     2048, 16, 32, 30, 5);
  bn_stats<<<32, 256, 0, stream>>>(ws + H1, ws + STATS, 2048, 32, 30);
  bn_apply<<<7680, 256, 0, stream>>>(ws + H1, ws + STATS, bne2_g, bne2_b,
                                     ws + H1, 2048, 32, 30, 960, 1);
  dec3_kernel<<<1024, 256, 0, stream>>>(ws + H1, dec3_w, dec3_b, out + XDEC);
}